// LLClusterCoordinates_5428838662733
// MI455X (gfx1250) — compile-verified
//
#include <hip/hip_runtime.h>
#include <math.h>

typedef __attribute__((ext_vector_type(2))) float v2f;
typedef __attribute__((ext_vector_type(8))) float v8f;

#define KSEG 512
#define QC   1.3017372f          /* atanh(0.5)^2 + Q_MIN */

/* workspace layout (float indices) */
#define WS_SUMX  0
#define WS_SUMY  512
#define WS_SUMZ  1024
#define WS_CNT   1536
#define WS_SATT  2048
#define WS_SREP  2049
#define WS_BMAT  2052            /* float4[512]: (-2c0,-2c1,-2c2,|c|^2)  (16B aligned: 2052*4=16*513) */
#define WS_CPACK 4100            /* float4[512]: (c0,c1,c2,qa)           (16B aligned: 4100*4=16*1025) */
#define WS_QA    6148            /* float[512] */

__global__ void k_zero(float* W) {
    int i = blockIdx.x * blockDim.x + threadIdx.x;
    if (i < 2050) W[i] = 0.0f;
}

/* Segment sums via LDS bins, then sparse global atomics */
__global__ void __launch_bounds__(256) k_accum(const float* __restrict__ coords,
                                               const int* __restrict__ tix,
                                               float* __restrict__ W, int n) {
    __shared__ float sx[KSEG], sy[KSEG], sz[KSEG], sc[KSEG];
    for (int k = threadIdx.x; k < KSEG; k += blockDim.x) { sx[k]=0.f; sy[k]=0.f; sz[k]=0.f; sc[k]=0.f; }
    __syncthreads();
    int i = blockIdx.x * blockDim.x + threadIdx.x;
    if (i < n) {
        int t = tix[i];
        if (t >= 0) {
            int s = t > (KSEG-1) ? (KSEG-1) : t;
            atomicAdd(&sx[s], coords[3*i+0]);
            atomicAdd(&sy[s], coords[3*i+1]);
            atomicAdd(&sz[s], coords[3*i+2]);
            atomicAdd(&sc[s], 1.0f);
        }
    }
    __syncthreads();
    for (int k = threadIdx.x; k < KSEG; k += blockDim.x) {
        if (sc[k] != 0.0f) {
            atomicAdd(&W[WS_SUMX + k], sx[k]);
            atomicAdd(&W[WS_SUMY + k], sy[k]);
            atomicAdd(&W[WS_SUMZ + k], sz[k]);
            atomicAdd(&W[WS_CNT  + k], sc[k]);
        }
    }
}

/* cc[k], qa[k], and WMMA-ready B operand table */
__global__ void k_fin(float* __restrict__ W) {
    int k = blockIdx.x * blockDim.x + threadIdx.x;
    if (k >= KSEG) return;
    float cnt   = W[WS_CNT + k];
    float denom = fmaxf(QC * cnt, 1e-6f);
    float c0 = QC * W[WS_SUMX + k] / denom;
    float c1 = QC * W[WS_SUMY + k] / denom;
    float c2 = QC * W[WS_SUMZ + k] / denom;
    float qa = (cnt > 0.0f) ? QC : 0.0f;
    ((float4*)(W + WS_BMAT ))[k] = make_float4(-2.0f*c0, -2.0f*c1, -2.0f*c2, c0*c0 + c1*c1 + c2*c2);
    ((float4*)(W + WS_CPACK))[k] = make_float4(c0, c1, c2, qa);
    W[WS_QA + k] = qa;
}

/* Pairwise kernel: 256 threads = 8 wave32; each wave owns a 16-point row tile.
 * d2(i,k)+eps = (A row (x,y,z,1)) . (B col (-2c0,-2c1,-2c2,|c|^2)) + C(|x_i|^2+eps)
 * — the |x|^2+eps add rides the WMMA C operand; epilogue = max, v_sqrt, sub, relu, fma. */
__global__ void __launch_bounds__(256) k_pair(const float* __restrict__ coords,
                                              const int* __restrict__ tix,
                                              float* __restrict__ W, int n) {
    __shared__ float4 bmS[KSEG];     /* 8 KB */
    __shared__ float  qaS[KSEG];     /* 2 KB */
    __shared__ float4 ptsS[128];     /* 2 KB: (x,y,z,|x|^2) */
    __shared__ float  blkAcc[2];     /* [0]=att, [1]=rep */

    const float4* bmG = (const float4*)(W + WS_BMAT);
    for (int k = threadIdx.x; k < KSEG; k += 256) { bmS[k] = bmG[k]; qaS[k] = W[WS_QA + k]; }

    int t = threadIdx.x;
    if (t < 128) {
        int i = blockIdx.x * 128 + t;
        float4 p;
        if (i < n) {
            float x = coords[3*i], y = coords[3*i+1], z = coords[3*i+2];
            p = make_float4(x, y, z, x*x + y*y + z*z);
        } else {
            p = make_float4(0.f, 0.f, 0.f, 1e12f);   /* pad rows: d2 huge -> relu = 0 */
        }
        ptsS[t] = p;
    }
    if (t == 0) { blkAcc[0] = 0.f; blkAcc[1] = 0.f; }
    __syncthreads();

    int wave = threadIdx.x >> 5;
    int lane = threadIdx.x & 31;
    int col  = lane & 15;
    bool lo  = (lane < 16);

    /* A operand (16x4 f32): row M = lane&15; VGPR0 holds K=0 (lo) / K=2 (hi), VGPR1 K=1 / K=3 */
    float4 pa = ptsS[wave * 16 + col];
    v2f a;
    a.x = lo ? pa.x : pa.z;
    a.y = lo ? pa.y : 1.0f;

    /* C operand: |x|^2 + eps for the 8 C/D rows this lane owns: M = 8*(lane>=16) + j */
    v8f xnv;
    int rbase = wave * 16 + (lo ? 0 : 8);
#pragma unroll
    for (int j = 0; j < 8; ++j) xnv[j] = ptsS[rbase + j].w + 1e-6f;

    float repAcc = 0.0f;
#pragma unroll 2
    for (int kt = 0; kt < KSEG / 16; ++kt) {
        int kg = kt * 16 + col;
        float4 bv = bmS[kg];
        v2f b;
        b.x = lo ? bv.x : bv.z;
        b.y = lo ? bv.y : bv.w;
        v8f acc = __builtin_amdgcn_wmma_f32_16x16x4_f32(
                      false, a, false, b, (short)0, xnv, false, false);
        float qa = qaS[kg];
#pragma unroll
        for (int j = 0; j < 8; ++j) {
            float d2 = fmaxf(acc[j], 0.0f);            /* guard rounding-negative */
            float s  = __builtin_amdgcn_sqrtf(d2);     /* bare v_sqrt_f32 (TRANS) */
            float r  = fmaxf(1.0f - s, 0.0f);
            repAcc   = fmaf(qa, r, repAcc);
        }
    }
    /* wave32 reduction */
    for (int off = 16; off > 0; off >>= 1) repAcc += __shfl_xor(repAcc, off, 32);
    if (lane == 0) atomicAdd(&blkAcc[1], repAcc);

    /* own-cluster attractive term and rep_own subtraction (one thread per point) */
    if (t < 128) {
        int i = blockIdx.x * 128 + t;
        if (i < n) {
            int tv = tix[i];
            if (tv >= 0) {
                int s = tv > (KSEG-1) ? (KSEG-1) : tv;
                float4 cp = ((const float4*)(W + WS_CPACK))[s];
                float4 p  = ptsS[t];
                float dx = p.x - cp.x, dy = p.y - cp.y, dz = p.z - cp.z;
                float d2o = dx*dx + dy*dy + dz*dz;
                float qa  = cp.w;
                float att = qa * d2o;
                float own = qa * fmaxf(1.0f - __builtin_amdgcn_sqrtf(d2o + 1e-6f), 0.0f);
                if (att != 0.0f) atomicAdd(&blkAcc[0], att);
                if (own != 0.0f) atomicAdd(&blkAcc[1], -own);
            }
        }
    }
    __syncthreads();
    if (threadIdx.x == 0) {
        atomicAdd(&W[WS_SATT], blkAcc[0]);
        atomicAdd(&W[WS_SREP], blkAcc[1]);
    }
}

__global__ void k_out(const float* __restrict__ W, float* __restrict__ out, int n) {
    /* loss = 0.5*V_att + 0.5*V_rep = 0.5*QC*(S_att + S_rep)/n */
    out[0] = 0.5f * QC * (W[WS_SATT] + W[WS_SREP]) / (float)n;
}

extern "C" void kernel_launch(void* const* d_in, const int* in_sizes, int n_in,
                              void* d_out, int out_size, void* d_ws, size_t ws_size,
                              hipStream_t stream) {
    const float* coords = (const float*)d_in[0];
    const int*   tix    = (const int*)d_in[1];
    int n = in_sizes[1];                 /* truth_indices is (N,1) -> flat count N */
    float* W   = (float*)d_ws;
    float* out = (float*)d_out;

    k_zero <<<9, 256, 0, stream>>>(W);
    k_accum<<<(n + 255) / 256, 256, 0, stream>>>(coords, tix, W, n);
    k_fin  <<<2, 256, 0, stream>>>(W);
    k_pair <<<(n + 127) / 128, 256, 0, stream>>>(coords, tix, W, n);
    k_out  <<<1, 1, 0, stream>>>(W, out, n);
}